// Node2AnchorSetAttentionUpdate_56023553409075
// MI455X (gfx1250) — compile-verified
//
#include <hip/hip_runtime.h>
#include <math.h>

// Problem constants (from reference)
#define H      256
#define HE     64
#define NB     8
#define A_PER  64
#define N_PER  512
#define NA     (NB * A_PER)   // 512 anchors total
#define NN     (NB * N_PER)   // 4096 nodes total

typedef __attribute__((ext_vector_type(2))) float v2f;
typedef __attribute__((ext_vector_type(8))) float v8f;

// ---------------------------------------------------------------------------
// fp32 WMMA GEMM, 64x32 output tile per wave32 (4x2 grid of 16x16 WMMA tiles).
//   C[b][m][n] = act( sum_k A[b][m][k]*B[b][k][n] + bias[n] (+C) )
//   B element addressed as Bbase + b*sBb + n*ldbn + k*ldbk (covers W^T / row-major)
// BCONTIG=true  -> ldbk==1, B fragment loaded as one aligned v2f (b64)
// BCONTIG=false -> arbitrary ldbk, two b32 loads per fragment
// Contiguous path: 8 v_wmma per 6 b64 loads. EXEC all-ones (blockDim=32).
// ---------------------------------------------------------------------------
struct GemmArgs {
  const float* A;  long lda;  long sAb;
  const float* B;  long ldbn; long ldbk; long sBb;
  float*       C;  long ldc;  long sCb;
  const float* bias;
  int K;
  int relu;
  int addC;
};

template <bool BCONTIG>
__global__ __launch_bounds__(32)
void wmma_gemm_f32(GemmArgs g) {
  const int lane = threadIdx.x;           // 0..31 (wave32)
  const int r15  = lane & 15;             // M row (A) / N col (B,C)
  const int kh   = (lane >> 4) << 1;      // lanes 0-15 -> K{0,1}, 16-31 -> K{2,3}
  const int tn = blockIdx.x, tm = blockIdx.y, bz = blockIdx.z;

  const float* Abase = g.A + (long)bz * g.sAb + (long)(tm * 64 + r15) * g.lda;
  const float* A0 = Abase;
  const float* A1 = Abase + (long)16 * g.lda;
  const float* A2 = Abase + (long)32 * g.lda;
  const float* A3 = Abase + (long)48 * g.lda;
  const float* B0 = g.B + (long)bz * g.sBb + (long)(tn * 32 + r15) * g.ldbn;
  const float* B1 = B0 + (long)16 * g.ldbn;

  const v8f zero = {0.f,0.f,0.f,0.f,0.f,0.f,0.f,0.f};
  v8f c00 = zero, c01 = zero, c10 = zero, c11 = zero;
  v8f c20 = zero, c21 = zero, c30 = zero, c31 = zero;

#if __has_builtin(__builtin_amdgcn_wmma_f32_16x16x4_f32)
  for (int k0 = 0; k0 < g.K; k0 += 4) {
    const int ka = k0 + kh;
    // A fragments: k-contiguous -> aligned v2f (global_load_b64)
    v2f a0 = *(const v2f*)(A0 + ka);
    v2f a1 = *(const v2f*)(A1 + ka);
    v2f a2 = *(const v2f*)(A2 + ka);
    v2f a3 = *(const v2f*)(A3 + ka);
    v2f b0, b1;
    if (BCONTIG) {
      b0 = *(const v2f*)(B0 + ka);
      b1 = *(const v2f*)(B1 + ka);
    } else {
      b0.x = B0[(long)ka * g.ldbk];  b0.y = B0[(long)(ka + 1) * g.ldbk];
      b1.x = B1[(long)ka * g.ldbk];  b1.y = B1[(long)(ka + 1) * g.ldbk];
    }
    // 8-arg VOP3P form: (neg_a, A, neg_b, B, c_mod, C, reuse_a, reuse_b)
    c00 = __builtin_amdgcn_wmma_f32_16x16x4_f32(false, a0, false, b0, (short)0, c00, false, false);
    c01 = __builtin_amdgcn_wmma_f32_16x16x4_f32(false, a0, false, b1, (short)0, c01, false, false);
    c10 = __builtin_amdgcn_wmma_f32_16x16x4_f32(false, a1, false, b0, (short)0, c10, false, false);
    c11 = __builtin_amdgcn_wmma_f32_16x16x4_f32(false, a1, false, b1, (short)0, c11, false, false);
    c20 = __builtin_amdgcn_wmma_f32_16x16x4_f32(false, a2, false, b0, (short)0, c20, false, false);
    c21 = __builtin_amdgcn_wmma_f32_16x16x4_f32(false, a2, false, b1, (short)0, c21, false, false);
    c30 = __builtin_amdgcn_wmma_f32_16x16x4_f32(false, a3, false, b0, (short)0, c30, false, false);
    c31 = __builtin_amdgcn_wmma_f32_16x16x4_f32(false, a3, false, b1, (short)0, c31, false, false);
  }
#else
  // Scalar fallback reproducing the WMMA C/D lane layout (diagnostic path only).
  {
    const float* Ab = g.A + (long)bz * g.sAb;
    const int mb = tm * 64 + ((lane >> 4) << 3);
    for (int k0 = 0; k0 < g.K; ++k0) {
      float bv0 = B0[(long)k0 * g.ldbk];
      float bv1 = B1[(long)k0 * g.ldbk];
      #pragma unroll
      for (int r = 0; r < 8; ++r) {
        float av0 = Ab[(long)(mb + r)      * g.lda + k0];
        float av1 = Ab[(long)(mb + 16 + r) * g.lda + k0];
        float av2 = Ab[(long)(mb + 32 + r) * g.lda + k0];
        float av3 = Ab[(long)(mb + 48 + r) * g.lda + k0];
        c00[r] += av0 * bv0;  c01[r] += av0 * bv1;
        c10[r] += av1 * bv0;  c11[r] += av1 * bv1;
        c20[r] += av2 * bv0;  c21[r] += av2 * bv1;
        c30[r] += av3 * bv0;  c31[r] += av3 * bv1;
      }
    }
  }
#endif

  // Epilogue: tile i covers rows tm*64 + i*16 + (lane>>4)*8 + r, cols tn*32+r15 (+16)
  const int  col0 = tn * 32 + r15;
  const int  col1 = col0 + 16;
  const int  m0   = tm * 64 + ((lane >> 4) << 3);
  const float bv0 = g.bias ? g.bias[col0] : 0.f;
  const float bv1 = g.bias ? g.bias[col1] : 0.f;
  float* Cp = g.C + (long)bz * g.sCb;
  const v8f* cc0[4] = { &c00, &c10, &c20, &c30 };
  const v8f* cc1[4] = { &c01, &c11, &c21, &c31 };
  #pragma unroll
  for (int i = 0; i < 4; ++i) {
    #pragma unroll
    for (int r = 0; r < 8; ++r) {
      const long row = (long)(m0 + i * 16 + r) * g.ldc;
      float v0 = (*cc0[i])[r] + bv0;
      float v1 = (*cc1[i])[r] + bv1;
      if (g.addC) { v0 += Cp[row + col0]; v1 += Cp[row + col1]; }
      if (g.relu) { v0 = fmaxf(v0, 0.f);  v1 = fmaxf(v1, 0.f); }
      Cp[row + col0] = v0;
      Cp[row + col1] = v1;
    }
  }
}

// ---------------------------------------------------------------------------
// Fused per-(b,a): distance -> RBF edge logits -> mask quirk -> softmax -> s
//   logit[n]  = Lbase[n] + sum_j qe[j]*exp(-(((d/10)-mu_j)/sigma)^2)
//   logit[n] *= (mask[n]-1)*1e6          (exact reference semantics)
//   attn      = softmax(logit)
//   s[j]      = sum_n attn[n]*rbf[n][j]  (rbf recomputed from cached dist)
// ---------------------------------------------------------------------------
__global__ __launch_bounds__(256)
void edge_softmax_kernel(const float* __restrict__ anchor_x,
                         const float* __restrict__ node_x,
                         const float* __restrict__ node_mask,
                         const float* __restrict__ qe,     // (NA, HE)
                         const float* __restrict__ Lbase,  // (NA, N_PER)
                         float* __restrict__ attn,         // (NA, N_PER)
                         float* __restrict__ s_out)        // (NA, HE)
{
  const int bid = blockIdx.x;      // b*64 + a
  const int b   = bid >> 6;
  const int t   = threadIdx.x;

  __shared__ float sh_qe[HE];
  __shared__ float sh_dist[N_PER];
  __shared__ float sh_l[N_PER];
  __shared__ float red[256];
  __shared__ float sh_bcast;
  __shared__ float sh_ax[3];

  if (t < HE) sh_qe[t] = qe[bid * HE + t];
  if (t < 3)  sh_ax[t] = anchor_x[bid * 3 + t];
  __syncthreads();

  const float axx = sh_ax[0], axy = sh_ax[1], axz = sh_ax[2];
  const float MU_STEP = 20.f / 63.f;      // linspace(0,20,64)
  const float INV_SIG = 64.f / 20.f;      // 1/sigma, sigma = 20/64

  for (int n = t; n < N_PER; n += 256) {
    const float* nx = node_x + (long)(b * N_PER + n) * 3;
    float dx = axx - nx[0] + 1e-8f;       // EPS added per-component (reference)
    float dy = axy - nx[1] + 1e-8f;
    float dz = axz - nx[2] + 1e-8f;
    float d  = sqrtf(dx * dx + dy * dy + dz * dz);
    sh_dist[n] = d;
    float d10 = d * 0.1f;
    float le = 0.f;
    #pragma unroll 8
    for (int j = 0; j < HE; ++j) {
      float u = (d10 - j * MU_STEP) * INV_SIG;
      le += sh_qe[j] * __expf(-u * u);
    }
    float lb = Lbase[(long)bid * N_PER + n] + le;
    lb *= (node_mask[b * N_PER + n] - 1.f) * 1000000.0f;   // reference's multiply quirk
    sh_l[n] = lb;
  }
  __syncthreads();

  // softmax over 512 entries with 256 threads
  red[t] = fmaxf(sh_l[t], sh_l[t + 256]);
  __syncthreads();
  for (int st = 128; st > 0; st >>= 1) {
    if (t < st) red[t] = fmaxf(red[t], red[t + st]);
    __syncthreads();
  }
  if (t == 0) sh_bcast = red[0];
  __syncthreads();
  const float mx = sh_bcast;

  float e0 = __expf(sh_l[t] - mx);
  float e1 = __expf(sh_l[t + 256] - mx);
  red[t] = e0 + e1;
  __syncthreads();
  for (int st = 128; st > 0; st >>= 1) {
    if (t < st) red[t] += red[t + st];
    __syncthreads();
  }
  if (t == 0) sh_bcast = red[0];
  __syncthreads();
  const float inv = 1.f / sh_bcast;

  const float a0 = e0 * inv, a1 = e1 * inv;
  sh_l[t] = a0; sh_l[t + 256] = a1;
  attn[(long)bid * N_PER + t]       = a0;
  attn[(long)bid * N_PER + t + 256] = a1;
  __syncthreads();

  // s[j] = sum_n attn[n]*rbf(dist[n], j), 64 j-values x 4 node-chunks
  const int j = t & 63, c = t >> 6;
  const float muj = j * MU_STEP;
  float part = 0.f;
  for (int n = c * 128; n < c * 128 + 128; ++n) {
    float u = (sh_dist[n] * 0.1f - muj) * INV_SIG;
    part += sh_l[n] * __expf(-u * u);
  }
  red[t] = part;
  __syncthreads();
  if (t < 64)
    s_out[bid * HE + t] = red[t] + red[t + 64] + red[t + 128] + red[t + 192];
}

// ---------------------------------------------------------------------------
// out[r] = LayerNorm(x[r] + y[r]) * g + b     (one 256-thread block per row)
// ---------------------------------------------------------------------------
__global__ __launch_bounds__(256)
void add_ln_kernel(const float* __restrict__ x, const float* __restrict__ y,
                   const float* __restrict__ gam, const float* __restrict__ bet,
                   float* __restrict__ out)
{
  const int r = blockIdx.x, t = threadIdx.x;
  __shared__ float red[256];
  __shared__ float sh_mean, sh_rstd;

  const float v = x[(long)r * H + t] + y[(long)r * H + t];
  red[t] = v;
  __syncthreads();
  for (int st = 128; st > 0; st >>= 1) { if (t < st) red[t] += red[t + st]; __syncthreads(); }
  if (t == 0) sh_mean = red[0] * (1.f / H);
  __syncthreads();

  const float d = v - sh_mean;
  red[t] = d * d;
  __syncthreads();
  for (int st = 128; st > 0; st >>= 1) { if (t < st) red[t] += red[t + st]; __syncthreads(); }
  if (t == 0) sh_rstd = rsqrtf(red[0] * (1.f / H) + 1e-5f);
  __syncthreads();

  out[(long)r * H + t] = d * sh_rstd * gam[t] + bet[t];
}

// ---------------------------------------------------------------------------
extern "C" void kernel_launch(void* const* d_in, const int* in_sizes, int n_in,
                              void* d_out, int out_size, void* d_ws, size_t ws_size,
                              hipStream_t stream) {
  (void)in_sizes; (void)n_in; (void)out_size; (void)ws_size;
  const float* anchor_x  = (const float*)d_in[0];
  const float* node_x    = (const float*)d_in[1];
  const float* anchor_f  = (const float*)d_in[2];
  const float* node_f    = (const float*)d_in[3];
  /* d_in[4] batch, d_in[5] anchor_batch: unused (regular batch layout) */
  const float* node_mask = (const float*)d_in[6];
  const float* Wq  = (const float*)d_in[7];
  const float* bq  = (const float*)d_in[8];
  const float* Wkv = (const float*)d_in[9];   // (512, 320) row-major
  const float* bkv = (const float*)d_in[10];
  const float* ln1_g = (const float*)d_in[11];
  const float* ln1_b = (const float*)d_in[12];
  const float* W1 = (const float*)d_in[13];   // (512, 256)
  const float* b1 = (const float*)d_in[14];
  const float* W2 = (const float*)d_in[15];   // (512, 512)
  const float* b2 = (const float*)d_in[16];
  const float* W3 = (const float*)d_in[17];   // (256, 512)
  const float* b3 = (const float*)d_in[18];
  const float* ln2_g = (const float*)d_in[19];
  const float* ln2_b = (const float*)d_in[20];

  const long WKV_LD = H + HE;                 // 320

  float* p = (float*)d_ws;
  float* q    = p; p += (long)NA * H;         // (512,256)
  float* qe   = p; p += (long)NA * HE;        // (512,64)
  float* kvn  = p; p += (long)NN * 2 * H;     // (4096,512)
  float* L    = p; p += (long)NA * N_PER;     // (512,512) base logits
  float* attn = p; p += (long)NA * N_PER;     // (512,512)
  float* sbuf = p; p += (long)NA * HE;        // (512,64)
  float* upd  = p; p += (long)NA * H;         // (512,256)
  float* af   = p; p += (long)NA * H;         // (512,256)
  float* m1   = p; p += (long)NA * 2 * H;     // (512,512)
  float* m2   = p; p += (long)NA * 2 * H;     // (512,512)
  float* m3   = p; p += (long)NA * H;         // (512,256)

  auto gemm = [&](const float* A, long lda, long sAb,
                  const float* Bm, long ldbn, long ldbk, long sBb,
                  float* C, long ldc, long sCb, const float* bias,
                  int M, int N, int K, int nb, int relu, int addC) {
    GemmArgs g{A, lda, sAb, Bm, ldbn, ldbk, sBb, C, ldc, sCb, bias, K, relu, addC};
    dim3 grid(N / 32, M / 64, nb);
    if (ldbk == 1)
      wmma_gemm_f32<true><<<grid, dim3(32), 0, stream>>>(g);
    else
      wmma_gemm_f32<false><<<grid, dim3(32), 0, stream>>>(g);
  };

  // 1) q = AF @ Wq^T + bq
  gemm(anchor_f, H, 0, Wq, H, 1, 0, q, H, 0, bq, NA, H, H, 1, 0, 0);
  // 2) qe = q @ We_k   (We_k[h][j] = Wkv[h][256+j])
  gemm(q, H, 0, Wkv + H, 1, WKV_LD, 0, qe, HE, 0, nullptr, NA, HE, H, 1, 0, 0);
  // 3) kvn = NF @ Wkv_n^T + bkv   (Wkv_n[c][h] = Wkv[c][h])
  gemm(node_f, H, 0, Wkv, WKV_LD, 1, 0, kvn, 2 * H, 0, bkv, NN, 2 * H, H, 1, 0, 0);
  // 4) base logits: L_b = q_b (64x256) @ k_n_b^T (256x512), batched over b
  gemm(q, H, (long)A_PER * H,
       kvn, 2 * H, 1, (long)N_PER * 2 * H,
       L, N_PER, (long)A_PER * N_PER, nullptr, A_PER, N_PER, H, NB, 0, 0);
  // 5) edge logits + mask quirk + softmax + s
  edge_softmax_kernel<<<dim3(NA), dim3(256), 0, stream>>>(
      anchor_x, node_x, node_mask, qe, L, attn, sbuf);
  // 6a) upd = attn_b (64x512) @ v_n_b (512x256), batched  (v_n = kvn[:,256:])
  gemm(attn, N_PER, (long)A_PER * N_PER,
       kvn + H, 1, 2 * H, (long)N_PER * 2 * H,
       upd, H, (long)A_PER * H, nullptr, A_PER, H, N_PER, NB, 0, 0);
  // 6b) upd += s @ We_v^T   (We_v[h][j] = Wkv[256+h][256+j])
  gemm(sbuf, HE, 0, Wkv + (long)H * WKV_LD + H, WKV_LD, 1, 0,
       upd, H, 0, nullptr, NA, H, HE, 1, 0, 1);
  // 7) af = LN(anchor_f + upd)
  add_ln_kernel<<<dim3(NA), dim3(256), 0, stream>>>(anchor_f, upd, ln1_g, ln1_b, af);
  // 8) m1 = relu(af @ W1^T + b1)
  gemm(af, H, 0, W1, H, 1, 0, m1, 2 * H, 0, b1, NA, 2 * H, H, 1, 1, 0);
  // 9) m2 = relu(m1 @ W2^T + b2)
  gemm(m1, 2 * H, 0, W2, 2 * H, 1, 0, m2, 2 * H, 0, b2, NA, 2 * H, 2 * H, 1, 1, 0);
  // 10) m3 = m2 @ W3^T + b3
  gemm(m2, 2 * H, 0, W3, 2 * H, 1, 0, m3, H, 0, b3, NA, H, 2 * H, 1, 0, 0);
  // 11) out = LN(af + m3)
  add_ln_kernel<<<dim3(NA), dim3(256), 0, stream>>>(af, m3, ln2_g, ln2_b, (float*)d_out);
}